// Horizontal_Conv_3D_77979426226522
// MI455X (gfx1250) — compile-verified
//
#include <hip/hip_runtime.h>

typedef __attribute__((ext_vector_type(2))) float v2f;
typedef __attribute__((ext_vector_type(8))) float v8f;

#define NL    10      // layers (also output innermost dim)
#define BB    8       // batch
#define CIN   64
#define COUT  128
#define HH    64
#define WWID  64
#define ZD    10      // input z extent
#define NEG_SLOPE 0.2f

#define WTILE 16              // spatial w positions per workgroup (GEMM M)
#define CIPAD 68              // padded ci stride in LDS (64 used) -> conflict-free b64
#define FILL_N (3*3*CIN*18)   // im2col tile elements: (dh,dz) x ci x w'

// ---- pre-pass: W[i][co][ci][d] -> Wt[i][d][ci/2][co][ci&1]  (d = dh*9+dw*3+dz)
// ci-pair interleaving makes each B fragment one contiguous 8-byte (b64) load.
__global__ void wt_transpose_kernel(const float* __restrict__ W,
                                    float* __restrict__ Wt) {
  int idx = blockIdx.x * 256 + threadIdx.x;
  const int total = NL * 27 * (CIN / 2) * COUT * 2;
  if (idx >= total) return;
  int par = idx & 1;
  int co  = (idx >> 1) & (COUT - 1);
  int t   = idx >> 8;              // ((i*27 + d) * 32 + cip)
  int cip = t & 31;
  t >>= 5;                         // i*27 + d
  int d = t % 27;
  int i = t / 27;
  Wt[idx] = W[(((size_t)(i * COUT + co) * CIN + cip * 2 + par) * 27) + d];
}

// ---- main implicit-GEMM conv using V_WMMA_F32_16X16X4_F32 ----
// A (16x4, per-lane rows)  = im2col(x) from LDS, M = spatial w, K = ci quad
// B (4x16, rows across lanes) = weights, N = co
template <bool TRANSPOSED>
__global__ __launch_bounds__(256)
void horiz_conv3d_wmma(const float* __restrict__ x,
                       const float* __restrict__ w,   // Wt if TRANSPOSED else W
                       float* __restrict__ out) {
  __shared__ float xs[9 * 18 * CIPAD];   // [dh*3+dz][w'][ci]  (ci innermost)
  __shared__ float tb[8 * 16 * 17];      // per-wave store-transpose tile

  const int w0 = blockIdx.x * WTILE;
  const int h  = blockIdx.y;
  const int bz = blockIdx.z;
  const int i  = bz >> 3;   // layer
  const int b  = bz & 7;

  const int tid    = threadIdx.x;
  const int lane   = tid & 31;
  const int wave   = tid >> 5;
  const int m      = lane & 15;   // N index (co) for B/D, M index (w) for A
  const int hi     = lane >> 4;   // half-wave select (K offset +2 / M offset +8)
  const int coBase = wave * 16;

  // ---- stage zero-padded im2col tile into LDS ----
  for (int t = tid; t < FILL_N; t += 256) {
    int wp = t % 18;
    int r  = t / 18;
    int ci = r & (CIN - 1);
    int q  = r >> 6;              // 0..8 = dh*3 + dz
    int dz = q % 3;
    int dh = q / 3;
    int hh = h  + dh - 1;
    int ww = w0 + wp - 1;
    int zz = i  + dz - 1;
    float v = 0.f;
    if ((unsigned)hh < (unsigned)HH && (unsigned)ww < (unsigned)WWID &&
        (unsigned)zz < (unsigned)ZD) {
      v = x[(((size_t)(b * CIN + ci) * HH + hh) * WWID + ww) * ZD + zz];
    }
    xs[(q * 18 + wp) * CIPAD + ci] = v;
  }
  __syncthreads();

  v8f acc0 = {0.f, 0.f, 0.f, 0.f, 0.f, 0.f, 0.f, 0.f};
  v8f acc1 = {0.f, 0.f, 0.f, 0.f, 0.f, 0.f, 0.f, 0.f};

  // Per-lane bases: everything loop-varying is a uniform constant offset.
  const float* xsl = &xs[m * CIPAD + hi * 2];          // + (q*18+dw)*CIPAD + ci0
  const v2f* wt2 = (const v2f*)w;                      // TRANSPOSED path
  const v2f* wl =
      wt2 + ((size_t)i * 27 * (CIN / 2)) * COUT + (size_t)hi * COUT + coBase + m;

  // K = 1728 reduced as 8 x (two interleaved ci-quads) x 27 taps -> 432 WMMAs
  for (int ci0 = 0; ci0 < CIN; ci0 += 8) {
#pragma unroll
    for (int dh = 0; dh < 3; ++dh) {
#pragma unroll
      for (int dw = 0; dw < 3; ++dw) {
#pragma unroll
        for (int dz = 0; dz < 3; ++dz) {
          const int q = dh * 3 + dz;
          const int d = dh * 9 + dw * 3 + dz;

          // A fragments: single aligned ds_load_b64 each (conflict-free)
          const v2f a0 = *(const v2f*)&xsl[(q * 18 + dw) * CIPAD + ci0];
          const v2f a1 = *(const v2f*)&xsl[(q * 18 + dw) * CIPAD + ci0 + 4];

          v2f b0, b1;
          if (TRANSPOSED) {
            // B fragments: single coalesced global_load_b64 each
            b0 = wl[(d * (CIN / 2) + (ci0 >> 1)) * COUT];
            b1 = wl[(d * (CIN / 2) + (ci0 >> 1) + 2) * COUT];
          } else {
            // fallback: scattered reads of original W[i][co][ci][d]
            const size_t bi =
                ((size_t)(i * COUT + coBase + m) * CIN + ci0 + hi * 2) * 27 + d;
            b0.x = w[bi];
            b0.y = w[bi + 27];
            b1.x = w[bi + 4 * 27];
            b1.y = w[bi + 5 * 27];
          }

          acc0 = __builtin_amdgcn_wmma_f32_16x16x4_f32(
              false, a0, false, b0, (short)0, acc0, false, false);
          acc1 = __builtin_amdgcn_wmma_f32_16x16x4_f32(
              false, a1, false, b1, (short)0, acc1, false, false);
        }
      }
    }
  }

  const v8f acc = acc0 + acc1;

  // ---- LeakyReLU + per-wave LDS transpose so stores walk w (coalesced-ish) ----
  float* tw = &tb[wave * (16 * 17)];
#pragma unroll
  for (int r = 0; r < 8; ++r) {
    float v = acc[r];                       // D: M = r + hi*8 (w), N = m (co)
    v = (v >= 0.f) ? v : v * NEG_SLOPE;
    tw[m * 17 + r + hi * 8] = v;            // tb[co_local][w_local]
  }
  __syncthreads();

  // out[b][co][h][w][i]
  const size_t outBase = (size_t)b * ((size_t)COUT * HH * WWID * NL) +
                         (size_t)h * (WWID * NL) + (size_t)w0 * NL + i;
#pragma unroll
  for (int cc = 0; cc < 8; ++cc) {
    const int col = cc + hi * 8;            // co_local
    const float v = tw[col * 17 + m];       // lane sweeps w_local = m
    out[outBase + (size_t)(coBase + col) * (HH * WWID * NL) + (size_t)m * NL] = v;
  }
}

extern "C" void kernel_launch(void* const* d_in, const int* in_sizes, int n_in,
                              void* d_out, int out_size, void* d_ws, size_t ws_size,
                              hipStream_t stream) {
  (void)in_sizes; (void)n_in; (void)out_size;
  const float* x = (const float*)d_in[0];
  const float* W = (const float*)d_in[1];
  float* out = (float*)d_out;

  const size_t wtElems = (size_t)NL * 27 * CIN * COUT;   // 2,211,840
  dim3 grid(WWID / WTILE, HH, NL * BB);                  // (4, 64, 80)

  if (ws_size >= wtElems * sizeof(float)) {
    float* Wt = (float*)d_ws;
    wt_transpose_kernel<<<(int)((wtElems + 255) / 256), 256, 0, stream>>>(W, Wt);
    horiz_conv3d_wmma<true><<<grid, 256, 0, stream>>>(x, Wt, out);
  } else {
    horiz_conv3d_wmma<false><<<grid, 256, 0, stream>>>(x, W, out);
  }
}